// dendriter_80152679678501
// MI455X (gfx1250) — compile-verified
//
#include <hip/hip_runtime.h>

#define UNITS 512
#define CONN  512
#define SEQL  32
#define BATCH 512

typedef __attribute__((ext_vector_type(2))) float v2f;
typedef __attribute__((ext_vector_type(8))) float v8f;

// ---------------------------------------------------------------------------
// Kernel 1: fold the per-unit scalar kernel weight and the gathered dendritic
// weight into one effective GEMM weight matrix:
//   W[u][c] = kernel[u] * dendriticW[ dendrites[u][c] ][u]
// Stored [UNITS][CONN] (i.e. B transposed) so GEMM B-fragments are contiguous.
// ---------------------------------------------------------------------------
__global__ __launch_bounds__(256) void dendriter_build_w(
    const float* __restrict__ kernelW,   // [UNITS]   (kernel[0][:])
    const float* __restrict__ dendW,     // [SEQL][UNITS]
    const int*   __restrict__ dend,      // [UNITS][CONN] segment ids
    float* __restrict__ W)               // [UNITS][CONN]
{
    const int idx = blockIdx.x * blockDim.x + threadIdx.x;  // 0..UNITS*CONN-1
    const int u   = idx >> 9;            // idx / CONN
    const int s   = dend[idx];           // coalesced (consecutive c)
    W[idx] = kernelW[u] * dendW[s * UNITS + u];
}

// ---------------------------------------------------------------------------
// Kernel 2: out = relu(x @ W^T + bias) via V_WMMA_F32_16X16X4_F32.
// 2x2 register blocking: one wave32 computes a 32x32 output tile with four
// independent accumulators. Per K-step (K=4): 2 A-frags + 2 B-frags feed
// 4 independent WMMAs -> 8 FLOP/byte from L2 and 4 parallel matrix chains.
//
// Fragment layouts (CDNA5 ISA 7.12.2, 32-bit matrices):
//   A (16x4, MxK): M = lane&15; VGPR v holds K = v + 2*(lane>>4)
//     -> per lane the 2 A values are consecutive floats of row M.
//   B (4x16, KxN): N = lane&15; VGPR v holds K = v + 2*(lane>>4)
//     -> with W stored [unit][conn], per lane 2 consecutive floats of row N.
//   C/D (16x16 f32): VGPR v: lanes 0-15 -> M=v, lanes 16-31 -> M=v+8; N=lane&15
// ---------------------------------------------------------------------------
__global__ __launch_bounds__(256) void dendriter_gemm(
    const float* __restrict__ x,     // [BATCH][CONN]
    const float* __restrict__ W,     // [UNITS][CONN]
    const float* __restrict__ bias,  // [UNITS]
    float* __restrict__ out)         // [BATCH][UNITS]
{
    const int lane = threadIdx.x & 31;
    const int wave = threadIdx.x >> 5;
    const int tile = blockIdx.x * 8 + wave;   // 256 super-tiles (32x32 each)
    const int bm   = tile >> 4;               // batch super-tile 0..15
    const int un   = tile & 15;               // unit  super-tile 0..15

    const int half = lane >> 4;               // K-half / M-half selector
    const int l    = lane & 15;               // M for A, N for B/C/D

    // Two A row-groups (batch rows l and l+16) and two B row-groups
    // (unit rows l and l+16) of the 32x32 super-tile.
    const float* arow0 = x + (size_t)(bm * 32 + l)      * CONN + 2 * half;
    const float* arow1 = x + (size_t)(bm * 32 + 16 + l) * CONN + 2 * half;
    const float* brow0 = W + (size_t)(un * 32 + l)      * CONN + 2 * half;
    const float* brow1 = W + (size_t)(un * 32 + 16 + l) * CONN + 2 * half;

    v8f acc00 = {};   // (M 0..15,  N 0..15)
    v8f acc01 = {};   // (M 0..15,  N 16..31)
    v8f acc10 = {};   // (M 16..31, N 0..15)
    v8f acc11 = {};   // (M 16..31, N 16..31)

#pragma unroll 4
    for (int k = 0; k < CONN; k += 4) {
        v2f a0 = *(const v2f*)(arow0 + k);    // global_load_b64
        v2f a1 = *(const v2f*)(arow1 + k);
        v2f b0 = *(const v2f*)(brow0 + k);
        v2f b1 = *(const v2f*)(brow1 + k);
        // Four independent accumulation chains.
        acc00 = __builtin_amdgcn_wmma_f32_16x16x4_f32(false, a0, false, b0,
                                                      (short)0, acc00, false, false);
        acc01 = __builtin_amdgcn_wmma_f32_16x16x4_f32(false, a0, false, b1,
                                                      (short)0, acc01, false, false);
        acc10 = __builtin_amdgcn_wmma_f32_16x16x4_f32(false, a1, false, b0,
                                                      (short)0, acc10, false, false);
        acc11 = __builtin_amdgcn_wmma_f32_16x16x4_f32(false, a1, false, b1,
                                                      (short)0, acc11, false, false);
    }

    // Epilogue: bias + ReLU, scatter the four 16x16 tiles.
    const float bs0 = bias[un * 32 + l];
    const float bs1 = bias[un * 32 + 16 + l];
    float* o00 = out + (size_t)(bm * 32 + half * 8)      * UNITS + un * 32 + l;
    float* o10 = out + (size_t)(bm * 32 + 16 + half * 8) * UNITS + un * 32 + l;
#pragma unroll
    for (int v = 0; v < 8; ++v) {
        float r00 = acc00[v] + bs0;
        float r01 = acc01[v] + bs1;
        float r10 = acc10[v] + bs0;
        float r11 = acc11[v] + bs1;
        o00[(size_t)v * UNITS]      = r00 > 0.0f ? r00 : 0.0f;
        o00[(size_t)v * UNITS + 16] = r01 > 0.0f ? r01 : 0.0f;
        o10[(size_t)v * UNITS]      = r10 > 0.0f ? r10 : 0.0f;
        o10[(size_t)v * UNITS + 16] = r11 > 0.0f ? r11 : 0.0f;
    }
}

// ---------------------------------------------------------------------------
// Host-side launcher.
// Inputs (setup_inputs order):
//   d_in[0] x          float [512*512]
//   d_in[1] kernel     float [512]          (shape [1,512] flattened)
//   d_in[2] dendriticW float [32*512]
//   d_in[3] bias       float [512]
//   d_in[4] dendrites  int32 [512*512]
// d_out: float [512*512]   d_ws: >= 1 MB scratch for W
// ---------------------------------------------------------------------------
extern "C" void kernel_launch(void* const* d_in, const int* in_sizes, int n_in,
                              void* d_out, int out_size, void* d_ws, size_t ws_size,
                              hipStream_t stream) {
    const float* x        = (const float*)d_in[0];
    const float* kernelW  = (const float*)d_in[1];
    const float* dendW    = (const float*)d_in[2];
    const float* bias     = (const float*)d_in[3];
    const int*   dend     = (const int*)d_in[4];
    float*       out      = (float*)d_out;
    float*       W        = (float*)d_ws;    // UNITS*CONN floats = 1 MB

    // Build effective weights: 512*512 elements, 256 threads/block.
    dendriter_build_w<<<(UNITS * CONN) / 256, 256, 0, stream>>>(kernelW, dendW, dend, W);

    // GEMM: 256 32x32 super-tiles, 8 waves (tiles) per 256-thread block.
    dendriter_gemm<<<(BATCH / 32) * (UNITS / 32) / 8, 256, 0, stream>>>(x, W, bias, out);
}